// GraphAttentionLayer_79139067396399
// MI455X (gfx1250) — compile-verified
//
#include <hip/hip_runtime.h>
#include <cstddef>
#include <cstdint>

// Problem constants (from reference setup_inputs)
constexpr int kB    = 8;
constexpr int kCin  = 64;
constexpr int kCout = 64;
constexpr int kN    = 2048;
constexpr int kT    = 13;
constexpr int kKt   = 3;
constexpr int kL    = 11;          // T - Kt + 1
constexpr int kJ    = kCout * kL;  // 704
constexpr float kAlpha = 0.2f;

typedef float v2f __attribute__((ext_vector_type(2)));
typedef float v8f __attribute__((ext_vector_type(8)));

// Low 32 bits of a generic pointer into LDS == LDS byte offset
// (flat aperture mapping keeps the low 32 address bits).
static __device__ __forceinline__ uint32_t lds_u32(const void* p) {
  return (uint32_t)(uintptr_t)p;
}

// CDNA5 async global->LDS DMA (ASYNCcnt-tracked), 16 bytes per lane.
static __device__ __forceinline__ void async_copy_b128(uint32_t lds_byte_addr,
                                                       const float* gaddr) {
  asm volatile("global_load_async_to_lds_b128 %0, %1, off"
               :
               : "v"(lds_byte_addr), "v"(gaddr)
               : "memory");
}

static __device__ __forceinline__ void wait_asynccnt0() {
  asm volatile("s_wait_asynccnt 0x0" ::: "memory");
}

// ---------------------------------------------------------------------------
// Kernel 1: conv0 (1xKt conv over T) + bias, writes H[b][n][j] (j = co*11+l)
// and f[b][n] = sum_j H*w1.  Block = 256 threads = 64(co) x 4(n).
// ---------------------------------------------------------------------------
__global__ __launch_bounds__(256) void gat_conv(
    const float* __restrict__ x,    // [B][Cin][N][T]
    const float* __restrict__ w0,   // [Cout][Cin][1][Kt]
    const float* __restrict__ b0,   // [Cout]
    const float* __restrict__ w1,   // [Cout*L]
    float* __restrict__ H,          // [B][N][J]
    float* __restrict__ F)          // [B][N]
{
  __shared__ float w0s[kCout * kCin * kKt];  // 12288 floats (48 KB)
  __shared__ float xs[kCin * 4 * kT];        // 3328 floats (13 KB), [ci][nn*13+t]

  const int bb  = blockIdx.y;
  const int n0  = blockIdx.x * 4;
  const int tid = threadIdx.x;

  for (int i = tid; i < kCout * kCin * kKt; i += 256) w0s[i] = w0[i];

  {  // stage x[b][ci][n0..n0+3][0..12]; 4 consecutive threads share one ci run
    const int ci = tid >> 2;
    const int p  = tid & 3;  // == nn
    const float* xp = x + ((size_t)(bb * kCin + ci) * kN + n0 + p) * kT;
    float* xd = &xs[ci * (4 * kT) + p * kT];
#pragma unroll
    for (int t = 0; t < kT; ++t) xd[t] = xp[t];
  }
  __syncthreads();

  const int co = tid & 63;
  const int nn = tid >> 6;
  const int n  = n0 + nn;

  float acc[kL];
  const float bias = b0[co];
#pragma unroll
  for (int l = 0; l < kL; ++l) acc[l] = bias;

  for (int ci = 0; ci < kCin; ++ci) {
    const float* xr = &xs[ci * (4 * kT) + nn * kT];
    const float* wr = &w0s[(co * kCin + ci) * kKt];
#pragma unroll
    for (int k = 0; k < kKt; ++k) {
      const float wk = wr[k];
#pragma unroll
      for (int l = 0; l < kL; ++l) acc[l] = fmaf(xr[l + k], wk, acc[l]);
    }
  }

  float* hr = H + ((size_t)bb * kN + n) * kJ + co * kL;
  float pf = 0.0f;
#pragma unroll
  for (int l = 0; l < kL; ++l) {
    hr[l] = acc[l];
    pf = fmaf(acc[l], w1[co * kL + l], pf);
  }

  // reduce pf over the 64 co-threads of each n (reuse xs as scratch)
  __syncthreads();
  xs[tid] = pf;
  __syncthreads();
  if (tid < 4) {
    float s = 0.0f;
    for (int i = 0; i < 64; ++i) s += xs[tid * 64 + i];
    F[(size_t)bb * kN + n0 + tid] = s;
  }
}

// ---------------------------------------------------------------------------
// Kernel 2a: per-row softmax stats over n of lrelu(f[q]+f[n]).
// One block per (b,q), 256 threads x 8 elements.
// ---------------------------------------------------------------------------
__global__ __launch_bounds__(256) void gat_rowstats(
    const float* __restrict__ F,
    float* __restrict__ RMAX,
    float* __restrict__ RINV)
{
  __shared__ float fs[kN];
  __shared__ float red[256];
  const int q   = blockIdx.x;
  const int bb  = blockIdx.y;
  const int tid = threadIdx.x;

  const float* Fb = F + (size_t)bb * kN;
  for (int i = tid; i < kN; i += 256) fs[i] = Fb[i];
  __syncthreads();

  const float fq = fs[q];
  float m = -3.4e38f;
#pragma unroll
  for (int i = 0; i < 8; ++i) {
    float z = fq + fs[tid + i * 256];
    z = (z >= 0.0f) ? z : kAlpha * z;
    m = fmaxf(m, z);
  }
  red[tid] = m;
  __syncthreads();
  for (int s = 128; s > 0; s >>= 1) {
    if (tid < s) red[tid] = fmaxf(red[tid], red[tid + s]);
    __syncthreads();
  }
  m = red[0];
  __syncthreads();

  float sum = 0.0f;
#pragma unroll
  for (int i = 0; i < 8; ++i) {
    float z = fq + fs[tid + i * 256];
    z = (z >= 0.0f) ? z : kAlpha * z;
    sum += __expf(z - m);
  }
  red[tid] = sum;
  __syncthreads();
  for (int s = 128; s > 0; s >>= 1) {
    if (tid < s) red[tid] += red[tid + s];
    __syncthreads();
  }
  if (tid == 0) {
    RMAX[(size_t)bb * kN + q] = m;
    RINV[(size_t)bb * kN + q] = 1.0f / red[0];
  }
}

// ---------------------------------------------------------------------------
// Kernel 2b: write attention directly in transposed layout att_T[b][n][q]
// (this is both the returned output and the A-operand source for the GEMM).
// 64x64 tiles; coalesced float4 stores along q.
// ---------------------------------------------------------------------------
__global__ __launch_bounds__(256) void gat_att(
    const float* __restrict__ F,
    const float* __restrict__ RMAX,
    const float* __restrict__ RINV,
    float* __restrict__ att)  // [B][N][N] = att_T[b][n][q]
{
  const int bb  = blockIdx.z;
  const int q0  = blockIdx.x * 64;
  const int n0  = blockIdx.y * 64;
  const int tid = threadIdx.x;
  const int qi  = (tid & 15) << 2;  // q offset 0..60 step 4
  const int nl  = tid >> 4;         // 0..15

  const size_t rb = (size_t)bb * kN;
  const float4 fq = *(const float4*)(F + rb + q0 + qi);
  const float4 mq = *(const float4*)(RMAX + rb + q0 + qi);
  const float4 iv = *(const float4*)(RINV + rb + q0 + qi);

#pragma unroll
  for (int p = 0; p < 4; ++p) {
    const int n = n0 + nl + p * 16;
    const float fn = F[rb + n];
    float4 v;
    float z;
    z = fq.x + fn; z = (z >= 0.0f) ? z : kAlpha * z; v.x = __expf(z - mq.x) * iv.x;
    z = fq.y + fn; z = (z >= 0.0f) ? z : kAlpha * z; v.y = __expf(z - mq.y) * iv.y;
    z = fq.z + fn; z = (z >= 0.0f) ? z : kAlpha * z; v.z = __expf(z - mq.z) * iv.z;
    z = fq.w + fn; z = (z >= 0.0f) ? z : kAlpha * z; v.w = __expf(z - mq.w) * iv.w;
    *(float4*)(att + (rb + n) * kN + q0 + qi) = v;
  }
}

// ---------------------------------------------------------------------------
// Kernel 3: h_prime[b][c][q][l] = sum_n att_T[b][n][q] * H[b][n][c*11+l]
// Per batch: [2048 x 2048] x [2048 x 704] fp32 GEMM via V_WMMA_F32_16X16X4_F32.
// Block = 4 waves; block tile 64(q) x 64(j); each wave owns 16(q) x 64(j).
// K chunked by 32; double-buffered LDS filled with ASYNCcnt-tracked
// global_load_async_to_lds_b128 DMAs issued ahead of the compute phase.
// ---------------------------------------------------------------------------
__global__ __launch_bounds__(128) void gat_gemm(
    const float* __restrict__ att,  // [B][N][N], att_T[b][n][q]
    const float* __restrict__ H,    // [B][N][J]
    float* __restrict__ out)        // [B][Cout][N][L]
{
  constexpr int PITCH = 80;         // conflict-free column fragment reads
  __shared__ float As[2][32 * PITCH];
  __shared__ float Bs[2][32 * PITCH];

  const int bb   = blockIdx.z;
  const int qb   = blockIdx.y * 64;
  const int jb   = blockIdx.x * 64;
  const int tid  = threadIdx.x;
  const int wv   = tid >> 5;
  const int lane = tid & 31;
  const int m16  = lane & 15;
  const int hi   = lane >> 4;  // half-wave: supplies K {0,1} (lo) or {2,3} (hi)

  const float* attB = att + (size_t)bb * kN * kN;
  const float* HB   = H + (size_t)bb * kN * kJ;

  // Per-thread staging slots: 4 x b128 per operand tile (32 rows x 64 cols)
  const int lin_r[4] = {(tid + 0) >> 4, (tid + 128) >> 4,
                        (tid + 256) >> 4, (tid + 384) >> 4};
  const int lin_c  = (tid & 15) << 2;

  v8f acc[4];
#pragma unroll
  for (int jt = 0; jt < 4; ++jt) acc[jt] = {};

  // Issue async DMAs for K-chunk kc into buffer p.
  auto stage = [&](int kc, int p) {
#pragma unroll
    for (int i = 0; i < 4; ++i) {
      const int r = lin_r[i];
      async_copy_b128(lds_u32(&As[p][r * PITCH + lin_c]),
                      attB + (size_t)(kc + r) * kN + qb + lin_c);
      async_copy_b128(lds_u32(&Bs[p][r * PITCH + lin_c]),
                      HB + (size_t)(kc + r) * kJ + jb + lin_c);
    }
  };

  stage(0, 0);
  wait_asynccnt0();
  __syncthreads();

  for (int kc = 0; kc < kN; kc += 32) {
    const int cur = (kc >> 5) & 1;
    if (kc + 32 < kN) stage(kc + 32, cur ^ 1);  // prefetch under the WMMAs

    const float* Ac = As[cur];
    const float* Bc = Bs[cur];
#pragma unroll
    for (int ks = 0; ks < 8; ++ks) {
      const int k0 = ks * 4 + 2 * hi;
      // A fragment (16x4 f32): lane m16 = row q; lo half K0/K1, hi half K2/K3
      v2f a;
      a.x = Ac[k0 * PITCH + wv * 16 + m16];
      a.y = Ac[(k0 + 1) * PITCH + wv * 16 + m16];
#pragma unroll
      for (int jt = 0; jt < 4; ++jt) {
        // B fragment (4x16 f32): lane m16 = col j; lo half rows 0/1, hi rows 2/3
        v2f bf;
        bf.x = Bc[k0 * PITCH + jt * 16 + m16];
        bf.y = Bc[(k0 + 1) * PITCH + jt * 16 + m16];
        acc[jt] = __builtin_amdgcn_wmma_f32_16x16x4_f32(
            false, a, false, bf, (short)0, acc[jt], false, false);
      }
    }

    wait_asynccnt0();   // my prefetches have landed
    __syncthreads();    // everyone done reading 'cur' and writing 'cur^1'
  }

  // D layout: VGPR r, lanes 0-15 -> M=r, lanes 16-31 -> M=r+8; col = lane%16
#pragma unroll
  for (int jt = 0; jt < 4; ++jt) {
    const int j  = jb + jt * 16 + m16;
    const int cc = j / kL;
    const int ll = j - cc * kL;
    float* op = out + ((size_t)(bb * kCout + cc) * kN) * kL + ll;
#pragma unroll
    for (int r = 0; r < 8; ++r) {
      const int q = qb + wv * 16 + hi * 8 + r;
      op[(size_t)q * kL] = acc[jt][r];
    }
  }
}

// ---------------------------------------------------------------------------
extern "C" void kernel_launch(void* const* d_in, const int* in_sizes, int n_in,
                              void* d_out, int out_size, void* d_ws, size_t ws_size,
                              hipStream_t stream) {
  const float* x  = (const float*)d_in[0];
  const float* w0 = (const float*)d_in[1];
  const float* b0 = (const float*)d_in[2];
  const float* w1 = (const float*)d_in[3];

  float* out    = (float*)d_out;
  float* hprime = out;                                       // [B][Cout][N][L]
  float* att    = out + (size_t)kB * kCout * kN * kL;        // [B][N][N]

  float* H    = (float*)d_ws;                                // [B][N][J]
  float* F    = H + (size_t)kB * kN * kJ;                    // [B][N]
  float* RMAX = F + (size_t)kB * kN;                         // [B][N]
  float* RINV = RMAX + (size_t)kB * kN;                      // [B][N]

  gat_conv<<<dim3(kN / 4, kB), 256, 0, stream>>>(x, w0, b0, w1, H, F);
  gat_rowstats<<<dim3(kN, kB), 256, 0, stream>>>(F, RMAX, RINV);
  gat_att<<<dim3(kN / 64, kN / 64, kB), 256, 0, stream>>>(F, RMAX, RINV, att);
  gat_gemm<<<dim3(kJ / 64, kN / 64, kB), 128, 0, stream>>>(att, H, hprime);
}